// MAMDense_87187836109346
// MI455X (gfx1250) — compile-verified
//
#include <hip/hip_runtime.h>
#include <cstdint>

// ---------------------------------------------------------------------------
// MAM dense (max-plus + min-plus semiring "GEMM" with argmax/argmin) for
// MI455X / gfx1250.  WMMA cannot express a max-reduction (and argmax needs a
// per-k compare regardless), so this is a register-tiled VALU kernel using the
// CDNA5 async global->LDS data path (global_load_async_to_lds_b32 +
// s_wait_asynccnt) with double buffering and b128 LDS reads.
//
//   C[m,n]    = max_k(x[m,k]*w[n,k]) + min_k(x[m,k]*w[n,k]) + bias[n]
//   amax[m,n] = argmax_k (first occurrence, matches jnp.argmax)
//   amin[m,n] = argmin_k
// ---------------------------------------------------------------------------

#define TM 64           // output tile rows per workgroup
#define TN 64           // output tile cols per workgroup
#define KT 32           // K slab staged in LDS per stage
#define LDS_STRIDE 68   // TM+4: keeps float4 reads 16B-aligned AND rotates banks
#define BUF_FLOATS (KT * LDS_STRIDE)   // one slab = 32 * 68 floats (8704 B)

__device__ __forceinline__ void async_ld_b32(uint32_t lds_off, uint32_t goff, uint64_t base) {
    // GVS form: global addr = 64-bit SGPR base + 32-bit VGPR offset; LDS dest addr taken from the VDST VGPR
    asm volatile("global_load_async_to_lds_b32 %0, %1, %2"
                 :: "v"(lds_off), "v"(goff), "s"(base) : "memory");
}
__device__ __forceinline__ void wait_async0() {
    asm volatile("s_wait_asynccnt 0" ::: "memory");
}

__global__ __launch_bounds__(256)
void mam_dense_kernel(const float* __restrict__ x,      // [M,K]
                      const float* __restrict__ w,      // [N,K]  (B = w^T)
                      const float* __restrict__ bias,   // [N]
                      float* __restrict__ out,          // [M,N]
                      int*   __restrict__ amax,         // [M,N]
                      int*   __restrict__ amin,         // [M,N]
                      int M, int N, int K)
{
    // Double-buffered transposed slabs: xs[k][r], ws[k][c], row stride 68.
    __shared__ __attribute__((aligned(16))) float xs[2 * BUF_FLOATS];
    __shared__ __attribute__((aligned(16))) float ws[2 * BUF_FLOATS];

    const int tid = threadIdx.x;
    const int tx  = tid & 15;          // 16 col-groups of 4
    const int ty  = tid >> 4;          // 16 row-groups of 4
    const int m0  = blockIdx.y * TM;
    const int n0  = blockIdx.x * TN;
    const int r0  = ty * 4;
    const int c0  = tx * 4;

    // Low 32 bits of the flat shared-memory pointer are the workgroup-relative
    // LDS byte offset.
    const uint32_t xs_base = (uint32_t)(uintptr_t)(&xs[0]);
    const uint32_t ws_base = (uint32_t)(uintptr_t)(&ws[0]);
    const uint64_t xg = (uint64_t)(uintptr_t)x;
    const uint64_t wg = (uint64_t)(uintptr_t)w;

    // Issue one slab's async loads (transposing scatter into LDS).
    // linear = it*256 + tid ; q = k-within-slab (lane-fast -> coalesced 128B
    // global reads), r = tile row/col.  LDS dest dword (q*68 + r): per wave
    // r is fixed, q = lane -> banks (4q + r) % 64, at worst 2-way on the
    // async write path (off the wave-critical path).
    auto issue = [&](int kbase, int bufsel) {
        const uint32_t xdst0 = xs_base + (uint32_t)bufsel * (BUF_FLOATS * 4u);
        const uint32_t wdst0 = ws_base + (uint32_t)bufsel * (BUF_FLOATS * 4u);
        #pragma unroll
        for (int it = 0; it < (TM * KT) / 256; ++it) {
            const int linear = it * 256 + tid;
            const int q = linear & (KT - 1);
            const int r = linear >> 5;
            const uint32_t ldsoff = (uint32_t)((q * LDS_STRIDE + r) * 4);
            const uint32_t gx = (uint32_t)(((m0 + r) * K + kbase + q) * 4);
            async_ld_b32(xdst0 + ldsoff, gx, xg);
            const uint32_t gw = (uint32_t)(((n0 + r) * K + kbase + q) * 4);
            async_ld_b32(wdst0 + ldsoff, gw, wg);
        }
    };

    // Accumulators: 4x4 outputs per thread, first-occurrence arg semantics.
    float cmax[4][4], cmin[4][4];
    int   imax[4][4], imin[4][4];
    #pragma unroll
    for (int i = 0; i < 4; ++i)
        #pragma unroll
        for (int j = 0; j < 4; ++j) {
            cmax[i][j] = -__builtin_inff();
            cmin[i][j] =  __builtin_inff();
            imax[i][j] = 0;
            imin[i][j] = 0;
        }

    const int NKT = K / KT;
    issue(0, 0);

    for (int kt = 0; kt < NKT; ++kt) {
        wait_async0();        // this wave's slab-kt asyncs landed in LDS
        __syncthreads();      // every wave's did; also fences buffer reuse

        if (kt + 1 < NKT) issue((kt + 1) * KT, (kt + 1) & 1);  // prefetch next slab

        const float* __restrict__ xb = &xs[(kt & 1) * BUF_FLOATS];
        const float* __restrict__ wb = &ws[(kt & 1) * BUF_FLOATS];
        const int kg0 = kt * KT;

        #pragma unroll 8
        for (int kk = 0; kk < KT; ++kk) {
            // 16B-aligned fragment loads -> single ds_load_b128 each.
            const float4 av = *(const float4*)(xb + kk * LDS_STRIDE + r0);
            const float4 bv = *(const float4*)(wb + kk * LDS_STRIDE + c0);
            const float a[4] = { av.x, av.y, av.z, av.w };
            const float b[4] = { bv.x, bv.y, bv.z, bv.w };
            const int kg = kg0 + kk;
            #pragma unroll
            for (int ii = 0; ii < 4; ++ii) {
                #pragma unroll
                for (int jj = 0; jj < 4; ++jj) {
                    const float p = a[ii] * b[jj];
                    const bool gm = p > cmax[ii][jj];   // strict: keep first k on ties
                    cmax[ii][jj] = gm ? p  : cmax[ii][jj];
                    imax[ii][jj] = gm ? kg : imax[ii][jj];
                    const bool lm = p < cmin[ii][jj];
                    cmin[ii][jj] = lm ? p  : cmin[ii][jj];
                    imin[ii][jj] = lm ? kg : imin[ii][jj];
                }
            }
        }
    }

    // Epilogue: C = cmax + cmin + bias, plus the two int index planes.
    const float4 bv = *(const float4*)(bias + n0 + c0);
    const float bb[4] = { bv.x, bv.y, bv.z, bv.w };

    #pragma unroll
    for (int ii = 0; ii < 4; ++ii) {
        const size_t row = (size_t)(m0 + r0 + ii) * (size_t)N + (size_t)(n0 + c0);
        float4 o;
        o.x = cmax[ii][0] + cmin[ii][0] + bb[0];
        o.y = cmax[ii][1] + cmin[ii][1] + bb[1];
        o.z = cmax[ii][2] + cmin[ii][2] + bb[2];
        o.w = cmax[ii][3] + cmin[ii][3] + bb[3];
        *(float4*)(out + row) = o;
        int4 ia = make_int4(imax[ii][0], imax[ii][1], imax[ii][2], imax[ii][3]);
        *(int4*)(amax + row) = ia;
        int4 ib = make_int4(imin[ii][0], imin[ii][1], imin[ii][2], imin[ii][3]);
        *(int4*)(amin + row) = ib;
    }
}

extern "C" void kernel_launch(void* const* d_in, const int* in_sizes, int n_in,
                              void* d_out, int out_size, void* d_ws, size_t ws_size,
                              hipStream_t stream) {
    const float* x    = (const float*)d_in[0];   // [M,K]
    const float* w    = (const float*)d_in[1];   // [N,K]
    const float* bias = (const float*)d_in[2];   // [N]

    const int N = in_sizes[2];
    const int K = in_sizes[1] / N;
    const int M = in_sizes[0] / K;

    float* out  = (float*)d_out;
    int*   amax = (int*)d_out + (size_t)M * N;
    int*   amin = (int*)d_out + 2 * (size_t)M * N;

    dim3 grid(N / TN, M / TM);   // 16 x 32 = 512 workgroups
    mam_dense_kernel<<<grid, 256, 0, stream>>>(x, w, bias, out, amax, amin, M, N, K);
}